// ClassifierGridSearch_69879117906561
// MI455X (gfx1250) — compile-verified
//
#include <hip/hip_runtime.h>
#include <hip/hip_bf16.h>
#include <stddef.h>

#define N_NODES 100000
#define N_EDGES 1600000
#define N_REL   16
#define IN_DIM  32
#define HID     64
#define NCLS    8
#define NGRAPH  128

#define NBINS   17                               // 16 relations + self-loop bin
#define EPAD    (N_EDGES + N_NODES + NBINS * 16) // binned edges, padded per bin
#define NTILES  ((EPAD + 15) / 16)
#define WPB     8                                // waves per block (256 thr)
#define MSG_BLOCKS ((NTILES + WPB - 1) / WPB)

typedef float v2f __attribute__((ext_vector_type(2)));
typedef float v8f __attribute__((ext_vector_type(8)));

// ---------------- small setup kernels ----------------

__global__ void rgcn_init_meta(int* __restrict__ counts,
                               float* __restrict__ pooled,
                               float* __restrict__ gcnt) {
  int t = threadIdx.x;
  if (t < NBINS) counts[t] = 0;
  for (int i = t; i < NGRAPH * HID; i += blockDim.x) pooled[i] = 0.0f;
  if (t < NGRAPH) gcnt[t] = 0.0f;
}

__global__ void rgcn_hist(const int* __restrict__ rel, int* __restrict__ counts) {
  int e = blockIdx.x * blockDim.x + threadIdx.x;
  if (e < N_EDGES) atomicAdd(&counts[rel[e]], 1);
}

__global__ void rgcn_prefix(const int* __restrict__ counts,
                            int* __restrict__ po, int* __restrict__ cursors,
                            int* __restrict__ bsrc, int* __restrict__ bdst) {
  if (threadIdx.x != 0 || blockIdx.x != 0) return;
  int off = 0;
  for (int r = 0; r < NBINS; ++r) {
    po[r] = off;
    cursors[r] = off;
    int c  = (r < N_REL) ? counts[r] : N_NODES;
    int cp = (c + 15) & ~15;
    for (int p = off + c; p < off + cp; ++p) { bsrc[p] = 0; bdst[p] = -1; }
    off += cp;
  }
  po[NBINS] = off;
}

__global__ void rgcn_bin_scatter(const int* __restrict__ src,
                                 const int* __restrict__ dst,
                                 const int* __restrict__ rel,
                                 int* __restrict__ cursors,
                                 int* __restrict__ bsrc, int* __restrict__ bdst) {
  int e = blockIdx.x * blockDim.x + threadIdx.x;
  if (e < N_EDGES) {
    int r = rel[e];
    int pos = atomicAdd(&cursors[r], 1);
    bsrc[pos] = src[e];
    bdst[pos] = dst[e];
  }
}

__global__ void rgcn_self_fill(const int* __restrict__ po,
                               int* __restrict__ bsrc, int* __restrict__ bdst) {
  int i = blockIdx.x * blockDim.x + threadIdx.x;
  if (i < N_NODES) {
    int pos = po[N_REL] + i;
    bsrc[pos] = i;
    bdst[pos] = i;
  }
}

__global__ void rgcn_bias_init(float* __restrict__ hv, const float* __restrict__ b) {
  int idx = blockIdx.x * blockDim.x + threadIdx.x;
  if (idx < N_NODES * HID) hv[idx] = b[idx & (HID - 1)];
}

__global__ void rgcn_relu(float* __restrict__ hv) {
  int idx = blockIdx.x * blockDim.x + threadIdx.x;
  if (idx < N_NODES * HID) hv[idx] = fmaxf(hv[idx], 0.0f);
}

// ---------------- WMMA message kernel ----------------
// One wave handles a tile of 16 binned edges, all sharing one relation.
// A (16xK f32) = gathered h[src] rows; B (Kx64) = W[rel]; 4 N-tiles of
// v_wmma_f32_16x16x4_f32; result rows scatter-added to hout[dst].
template <int K>
__global__ __launch_bounds__(256)
void rgcn_msg(const float* __restrict__ hin,   // [N,K]
              const float* __restrict__ Wrel,  // [16,K,HID]
              const float* __restrict__ Wloop, // [K,HID]
              float* __restrict__ hout,        // [N,HID] (pre-seeded with bias)
              const int* __restrict__ bsrc,
              const int* __restrict__ bdst,
              const int* __restrict__ po) {    // [NBINS+1]
  const int lane = threadIdx.x & 31;
  const int wave = threadIdx.x >> 5;
  const int tile = blockIdx.x * (blockDim.x >> 5) + wave;
  const int p0   = tile * 16;
  if (p0 >= po[NBINS]) return;               // wave-uniform exit

  int relid = 0;
  #pragma unroll
  for (int r = 1; r < NBINS; ++r) relid += (p0 >= po[r]) ? 1 : 0;
  const float* W = (relid == N_REL) ? Wloop : (Wrel + (size_t)relid * K * HID);

  // A fragment source: lane L holds row M=L%16, K pair {2*(L/16), 2*(L/16)+1}
  const int m     = lane & 15;
  const int khalf = lane >> 4;
  const int col   = lane & 15;               // B/C/D: N = lane%16
  const int s0    = bsrc[p0 + m];
  const float* arow = hin + (size_t)s0 * K + 2 * khalf;

  v8f c0 = {}, c1 = {}, c2 = {}, c3 = {};
  #pragma unroll
  for (int s = 0; s < K / 4; ++s) {
    v2f a;
    a.x = arow[4 * s + 0];
    a.y = arow[4 * s + 1];
    const float* wp = W + (size_t)(4 * s + 2 * khalf) * HID + col;
    v2f b0, b1, b2, b3;
    b0.x = wp[0];      b0.y = wp[HID];
    b1.x = wp[16];     b1.y = wp[HID + 16];
    b2.x = wp[32];     b2.y = wp[HID + 32];
    b3.x = wp[48];     b3.y = wp[HID + 48];
    c0 = __builtin_amdgcn_wmma_f32_16x16x4_f32(false, a, false, b0, (short)0, c0, false, false);
    c1 = __builtin_amdgcn_wmma_f32_16x16x4_f32(false, a, false, b1, (short)0, c1, false, false);
    c2 = __builtin_amdgcn_wmma_f32_16x16x4_f32(false, a, false, b2, (short)0, c2, false, false);
    c3 = __builtin_amdgcn_wmma_f32_16x16x4_f32(false, a, false, b3, (short)0, c3, false, false);
  }

  // C/D layout: reg j holds M = j + 8*(lane/16), N = 16*ntile + lane%16
  #pragma unroll
  for (int j = 0; j < 8; ++j) {
    const int M = j + 8 * khalf;
    const int d = bdst[p0 + M];
    if (d >= 0) {
      float* orow = hout + (size_t)d * HID + col;
      atomicAdd(orow + 0,  c0[j]);
      atomicAdd(orow + 16, c1[j]);
      atomicAdd(orow + 32, c2[j]);
      atomicAdd(orow + 48, c3[j]);
    }
  }
}

// ---------------- pooling + classifier ----------------

__global__ void rgcn_pool(const float* __restrict__ h2,
                          const int* __restrict__ n2g,
                          float* __restrict__ pooled, float* __restrict__ gcnt) {
  int idx = blockIdx.x * blockDim.x + threadIdx.x;
  if (idx >= N_NODES * HID) return;
  int n = idx >> 6;
  int o = idx & (HID - 1);
  int g = n2g[n];
  float v = fmaxf(h2[idx], 0.0f);            // ReLU fused with pooling
  atomicAdd(&pooled[g * HID + o], v);
  if (o == 0) atomicAdd(&gcnt[g], 1.0f);
}

__global__ void rgcn_classify(const float* __restrict__ pooled,
                              const float* __restrict__ gcnt,
                              const float* __restrict__ Wc,  // [HID, NCLS]
                              const float* __restrict__ bc,
                              float* __restrict__ out) {     // [NGRAPH, NCLS]
  int g = threadIdx.x;
  if (g >= NGRAPH) return;
  float inv = 1.0f / fmaxf(gcnt[g], 1.0f);
  float logits[NCLS];
  #pragma unroll
  for (int o = 0; o < NCLS; ++o) logits[o] = bc[o];
  for (int k = 0; k < HID; ++k) {
    float v = pooled[g * HID + k] * inv;
    #pragma unroll
    for (int o = 0; o < NCLS; ++o) logits[o] += v * Wc[k * NCLS + o];
  }
  float mx = logits[0];
  #pragma unroll
  for (int o = 1; o < NCLS; ++o) mx = fmaxf(mx, logits[o]);
  float sum = 0.0f;
  #pragma unroll
  for (int o = 0; o < NCLS; ++o) { logits[o] = expf(logits[o] - mx); sum += logits[o]; }
  float is = 1.0f / sum;
  #pragma unroll
  for (int o = 0; o < NCLS; ++o) out[g * NCLS + o] = logits[o] * is;
}

// ---------------- launcher ----------------

extern "C" void kernel_launch(void* const* d_in, const int* in_sizes, int n_in,
                              void* d_out, int out_size, void* d_ws, size_t ws_size,
                              hipStream_t stream) {
  const float* h   = (const float*)d_in[0];
  const int*   src = (const int*)d_in[1];
  const int*   dst = (const int*)d_in[2];
  const int*   rel = (const int*)d_in[3];
  const int*   n2g = (const int*)d_in[4];
  const float* W1  = (const float*)d_in[5];
  const float* Wl1 = (const float*)d_in[6];
  const float* b1  = (const float*)d_in[7];
  const float* W2  = (const float*)d_in[8];
  const float* Wl2 = (const float*)d_in[9];
  const float* b2  = (const float*)d_in[10];
  const float* Wc  = (const float*)d_in[11];
  const float* bc  = (const float*)d_in[12];
  float* out = (float*)d_out;

  char* wsb = (char*)d_ws;
  size_t off = 0;
  auto take = [&](size_t bytes) -> char* {
    char* p = wsb + off;
    off = (off + bytes + 255) & ~(size_t)255;
    return p;
  };
  float* h1      = (float*)take((size_t)N_NODES * HID * 4);
  float* h2      = (float*)take((size_t)N_NODES * HID * 4);
  int*   bsrc    = (int*)take((size_t)EPAD * 4);
  int*   bdst    = (int*)take((size_t)EPAD * 4);
  int*   counts  = (int*)take(64 * 4);
  int*   cursors = (int*)take(64 * 4);
  int*   po      = (int*)take(64 * 4);
  float* pooled  = (float*)take((size_t)NGRAPH * HID * 4);
  float* gcnt    = (float*)take((size_t)NGRAPH * 4);

  const int NT = 256;
  rgcn_init_meta<<<1, NT, 0, stream>>>(counts, pooled, gcnt);
  rgcn_hist<<<(N_EDGES + NT - 1) / NT, NT, 0, stream>>>(rel, counts);
  rgcn_prefix<<<1, 1, 0, stream>>>(counts, po, cursors, bsrc, bdst);
  rgcn_bin_scatter<<<(N_EDGES + NT - 1) / NT, NT, 0, stream>>>(src, dst, rel, cursors, bsrc, bdst);
  rgcn_self_fill<<<(N_NODES + NT - 1) / NT, NT, 0, stream>>>(po, bsrc, bdst);

  // layer 1
  rgcn_bias_init<<<(N_NODES * HID + NT - 1) / NT, NT, 0, stream>>>(h1, b1);
  rgcn_msg<IN_DIM><<<MSG_BLOCKS, NT, 0, stream>>>(h, W1, Wl1, h1, bsrc, bdst, po);
  rgcn_relu<<<(N_NODES * HID + NT - 1) / NT, NT, 0, stream>>>(h1);

  // layer 2
  rgcn_bias_init<<<(N_NODES * HID + NT - 1) / NT, NT, 0, stream>>>(h2, b2);
  rgcn_msg<HID><<<MSG_BLOCKS, NT, 0, stream>>>(h1, W2, Wl2, h2, bsrc, bdst, po);

  // ReLU + mean-pool + classify
  rgcn_pool<<<(N_NODES * HID + NT - 1) / NT, NT, 0, stream>>>(h2, n2g, pooled, gcnt);
  rgcn_classify<<<1, NGRAPH, 0, stream>>>(pooled, gcnt, Wc, bc, out);
}